// YoloLoss_33234456936690
// MI455X (gfx1250) — compile-verified
//
#include <hip/hip_runtime.h>

#define YG 7              // YOLO grid
#define YEPS 1e-6f

typedef __attribute__((ext_vector_type(2))) float v2f;
typedef __attribute__((ext_vector_type(8))) float v8f;

__device__ __forceinline__ float sigm(float x) {
    // 1 / (1 + e^-x); v_rcp_f32 + fast exp, ~1 ulp — fine for a loss sum
    return __builtin_amdgcn_rcpf(1.0f + __expf(-x));
}

// ---------------------------------------------------------------------------
// Exact fp32 wave32 sum on the matrix pipe: 5x V_WMMA_F32_16X16X4_F32.
// Stage 1: A = lane values (A[m][0]=x[m], A[m][2]=x[m+16], K1/K3=0), B = ones
//          -> D1[m][n] = s_m = x[m] + x[m+16] for every column n.
// Stage 2: A = ones, B = {D1 vgpr pairs}; with A all-ones the WMMA computes a
//          pure column sum, and column n of B lives in lanes {n, n+16} of its
//          two VGPRs, so 4 accumulating WMMAs gather s_0..s_15 exactly once.
// Result: every lane of acc holds sum over all 32 lanes, full fp32.
// Requires EXEC all-ones (call from converged full blocks only).
// ---------------------------------------------------------------------------
__device__ __forceinline__ float wave_sum32(float x) {
    v2f a;   a[0] = x;    a[1] = 0.0f;
    v2f one; one[0] = 1.0f; one[1] = 1.0f;
    v8f z = {};
    v8f d1 = __builtin_amdgcn_wmma_f32_16x16x4_f32(false, a, false, one,
                                                   (short)0, z, false, false);
    v8f acc = {};
    v2f b;
    b[0] = d1[0]; b[1] = d1[1];
    acc = __builtin_amdgcn_wmma_f32_16x16x4_f32(false, one, false, b,
                                                (short)0, acc, false, false);
    b[0] = d1[2]; b[1] = d1[3];
    acc = __builtin_amdgcn_wmma_f32_16x16x4_f32(false, one, false, b,
                                                (short)0, acc, false, false);
    b[0] = d1[4]; b[1] = d1[5];
    acc = __builtin_amdgcn_wmma_f32_16x16x4_f32(false, one, false, b,
                                                (short)0, acc, false, false);
    b[0] = d1[6]; b[1] = d1[7];
    acc = __builtin_amdgcn_wmma_f32_16x16x4_f32(false, one, false, b,
                                                (short)0, acc, false, false);
    return acc[0];
}

// Block (256 = 8 waves) sum; valid on threadIdx.x == 0.
__device__ __forceinline__ float block_sum256(float x) {
    __shared__ float wsum[8];
    float w = wave_sum32(x);
    const int lane = threadIdx.x & 31;
    const int wid  = threadIdx.x >> 5;
    if (lane == 0) wsum[wid] = w;
    __syncthreads();
    float s = 0.0f;
    if (threadIdx.x == 0) {
#pragma unroll
        for (int i = 0; i < 8; ++i) s += wsum[i];
    }
    return s;
}

__device__ __forceinline__ float iou_f(float pcx, float pcy, float pw, float ph,
                                       float tcx, float tcy, float tw, float th) {
    const float pl = pcx - pw * 0.5f, pr = pcx + pw * 0.5f;
    const float pt = pcy - ph * 0.5f, pb = pcy + ph * 0.5f;
    const float tl = tcx - tw * 0.5f, tr = tcx + tw * 0.5f;
    const float tt = tcy - th * 0.5f, tb = tcy + th * 0.5f;
    float ix = fminf(pr, tr) - fmaxf(pl, tl); ix = fmaxf(ix, 0.0f);
    float iy = fminf(pb, tb) - fmaxf(pt, tt); iy = fmaxf(iy, 0.0f);
    const float ia = ix * iy;
    const float pa = (pr - pl) * (pb - pt);   // = pw*ph (>=0)
    const float ta = (tr - tl) * (tb - tt);   // = tw*th (>=0)
    return ia / (pa + ta - ia + YEPS);
}

__global__ void __launch_bounds__(256)
yolo_loss_partial(const float* __restrict__ preds,
                  const float* __restrict__ targets,
                  float* __restrict__ partial, int ncells) {
    const int stride = gridDim.x * 256;
    float acc = 0.0f;

    for (int c = blockIdx.x * 256 + threadIdx.x; c < ncells; c += stride) {
        // prefetch the next tile this thread will touch (global_prefetch_b8)
        const int nxt = c + stride;
        if (nxt < ncells) {
            __builtin_prefetch(preds + (size_t)nxt * 12, 0, 1);
            __builtin_prefetch(targets + (size_t)nxt * 7, 0, 1);
        }

        const float* p = preds   + (size_t)c * 12;   // 48B-aligned
        const float* t = targets + (size_t)c * 7;
        const float4 pA = *(const float4*)(p);       // p0..p3
        const float4 pB = *(const float4*)(p + 4);   // p4..p7
        const float4 pC = *(const float4*)(p + 8);   // p8..p11
        const float t0 = t[0], t1 = t[1], t2 = t[2], t3 = t[3];
        const float t4 = t[4], t5 = t[5], t6 = t[6];

        const int   g     = c % (YG * YG);
        const float cellx = (float)(g % YG);
        const float celly = (float)(g / YG);
        const float inv7  = 1.0f / (float)YG;

        // decode pred box 1 (p0..p4)
        const float b1cx = (sigm(pA.x) + cellx) * inv7;
        const float b1cy = (sigm(pA.y) + celly) * inv7;
        float q = sigm(pA.z); const float b1w = q * q;
        q = sigm(pA.w);       const float b1h = q * q;
        const float b1cf = sigm(pB.x);
        // decode pred box 2 (p5..p9)
        const float b2cx = (sigm(pB.y) + cellx) * inv7;
        const float b2cy = (sigm(pB.z) + celly) * inv7;
        q = sigm(pB.w);       const float b2w = q * q;
        q = sigm(pC.x);       const float b2h = q * q;
        const float b2cf = sigm(pC.y);
        // decode target (t0..t4)
        const float tcx = (t0 + cellx) * inv7;
        const float tcy = (t1 + celly) * inv7;
        const float tw  = t2 * t2;
        const float th  = t3 * t3;

        const float iou1 = iou_f(b1cx, b1cy, b1w, b1h, tcx, tcy, tw, th);
        const float iou2 = iou_f(b2cx, b2cy, b2w, b2h, tcx, tcy, tw, th);
        const bool  sel  = iou1 > iou2;

        const float pcx = sel ? b1cx : b2cx;
        const float pcy = sel ? b1cy : b2cy;
        const float pw  = sel ? b1w  : b2w;
        const float ph  = sel ? b1h  : b2h;
        const float pcf = sel ? b1cf : b2cf;
        const float ior = sel ? iou1 : iou2;

        const float obj = (t4 > 0.0f) ? 1.0f : 0.0f;

        const float dx = pcx - tcx, dy = pcy - tcy;
        const float xy = dx * dx + dy * dy;

        const float sw = sqrtf(fmaxf(pw, YEPS)) - fabsf(t2);  // sqrt(t2^2)=|t2|
        const float sh = sqrtf(fmaxf(ph, YEPS)) - fabsf(t3);
        const float wh = sw * sw + sh * sh;

        const float dcf     = pcf - ior;
        const float confobj = dcf * dcf;
        const float noobj   = (1.0f - obj) * pcf * pcf;

        // 2-class softmax on p10,p11
        const float mx = fmaxf(pC.z, pC.w);
        const float e0 = __expf(pC.z - mx);
        const float e1 = __expf(pC.w - mx);
        const float is = __builtin_amdgcn_rcpf(e0 + e1);
        const float d0 = e0 * is - t5;
        const float d1v = e1 * is - t6;
        const float cls = d0 * d0 + d1v * d1v;

        acc += obj * (5.0f * (xy + wh) + confobj + cls) + 0.5f * noobj;
    }

    const float bs = block_sum256(acc);
    if (threadIdx.x == 0) partial[blockIdx.x] = bs;
}

__global__ void __launch_bounds__(256)
yolo_loss_finalize(const float* __restrict__ partial,
                   float* __restrict__ out, int n) {
    float acc = 0.0f;
    for (int i = threadIdx.x; i < n; i += 256) acc += partial[i];
    const float s = block_sum256(acc);
    if (threadIdx.x == 0) out[0] = s;
}

extern "C" void kernel_launch(void* const* d_in, const int* in_sizes, int n_in,
                              void* d_out, int out_size, void* d_ws, size_t ws_size,
                              hipStream_t stream) {
    const float* preds   = (const float*)d_in[0];
    const float* targets = (const float*)d_in[1];
    float* out     = (float*)d_out;
    float* partial = (float*)d_ws;

    const int ncells = in_sizes[0] / 12;            // B * 7 * 7

    // ~12 cells per thread; 244MB stream across ~268K threads
    int blocks = (ncells + (256 * 12) - 1) / (256 * 12);
    if (blocks > 2048) blocks = 2048;
    if ((size_t)blocks * sizeof(float) > ws_size) {
        size_t m = ws_size / sizeof(float);
        blocks = (m > 0) ? (int)m : 1;
    }
    if (blocks < 1) blocks = 1;

    yolo_loss_partial<<<blocks, 256, 0, stream>>>(preds, targets, partial, ncells);
    yolo_loss_finalize<<<1, 256, 0, stream>>>(partial, out, blocks);
}